// LearnableEdgeEmbeddings_4698694222002
// MI455X (gfx1250) — compile-verified
//
#include <hip/hip_runtime.h>
#include <math.h>

// CDNA5 wave32 WMMA operand types
typedef __attribute__((ext_vector_type(2))) float v2f;
typedef __attribute__((ext_vector_type(8))) float v8f;

// ---------------------------------------------------------------------------
// Kernel 1: zero the 256MB output with wide b128 stores (bandwidth floor).
// ---------------------------------------------------------------------------
__global__ void zero_out_kernel(float4* __restrict__ out4, long long n4,
                                float* __restrict__ out, long long n) {
    long long i = (long long)blockIdx.x * blockDim.x + threadIdx.x;
    long long stride = (long long)gridDim.x * blockDim.x;
    float4 z; z.x = 0.f; z.y = 0.f; z.z = 0.f; z.w = 0.f;
    for (long long k = i; k < n4; k += stride) out4[k] = z;
    // tail (out_size not multiple of 4) — empty for 1024*1024*64
    for (long long k = n4 * 4 + i; k < n; k += stride) out[k] = 0.f;
}

// ---------------------------------------------------------------------------
// Kernel 2: WMMA one-hot gather + atomic scatter.
// Per wave, per batch of 16 edges:
//   A[16x16] : A[e][k] = (k == et_e) ? val_e : 0   (one-hot over all 16 types)
//   B[16x64] : embedding table (batch-invariant, hoisted into VGPRs)
//   D = A x B  via V_WMMA_F32_16X16X4_F32 (4 K-steps x 4 N-tiles)
//   => D[e][n] == val_e * emb[et_e][n] exactly (single nonzero product term).
// Route row D[e] to out[src,dst,:] (et<8) or out[dst,src,:] (et>=8) with
// global_atomic_add_f32; the 4 N-tiles share one base address (imm offsets).
// ---------------------------------------------------------------------------
__global__ __launch_bounds__(256)
void edge_wmma_scatter(const int*   __restrict__ edge_type,   // (E,1)
                       const int2*  __restrict__ edge_out,    // (E,2) src,dst
                       const float* __restrict__ edge_val,    // (E)
                       const float* __restrict__ emb,         // (16,64)
                       float*       __restrict__ out,         // (N,N,64)
                       int E, int N) {
    __shared__ float ldsEmb[16 * 64];   // 4KB embedding table

    const int tid = threadIdx.x;

    // Stage 16x64 f32 table into LDS with b128 loads/stores.
    {
        const float4* s = (const float4*)emb;
        float4*       d = (float4*)ldsEmb;
        for (int i = tid; i < (16 * 64) / 4; i += blockDim.x) d[i] = s[i];
    }
    __syncthreads();

    const int lane  = tid & 31;
    const int half  = lane >> 4;      // 0: lanes 0-15, 1: lanes 16-31
    const int l16   = lane & 15;
    const int wid   = (blockIdx.x * (blockDim.x >> 5)) + (tid >> 5);
    const int nwav  = gridDim.x * (blockDim.x >> 5);
    const int nbat  = (E + 15) >> 4;

    // Hoist batch-invariant B operands into VGPRs.
    // ISA 32-bit B 4x16 layout: lane = N; VGPR{0,1} hold rows K={0,1}
    // (lanes 0-15) / K={2,3} (lanes 16-31).
    v2f Bop[4][4];   // [K-step s][N-tile nt]
#pragma unroll
    for (int s = 0; s < 4; ++s) {
        const int kk = s * 4 + half * 2;
#pragma unroll
        for (int nt = 0; nt < 4; ++nt) {
            Bop[s][nt].x = ldsEmb[kk * 64       + nt * 16 + l16];
            Bop[s][nt].y = ldsEmb[(kk + 1) * 64 + nt * 16 + l16];
        }
    }

    for (int b = wid; b < nbat; b += nwav) {
        const int  e     = b * 16 + l16;      // this lane's edge (both halves)
        const bool valid = (e < E);
        const int   et   = valid ? edge_type[e] : 0;
        const int2  sd   = valid ? edge_out[e]  : make_int2(0, 0);
        const float val  = valid ? edge_val[e]  : 0.0f;

        // Per-lane routed (row*N + col) for this lane's edge; fits in 32 bits.
        const int row    = (et < 8) ? sd.x : sd.y;
        const int col    = (et < 8) ? sd.y : sd.x;
        const int rowcol = row * N + col;

        // One-hot A for the 4 K-steps.
        // ISA 32-bit A 16x4 layout: lane = M; VGPR{0,1} hold K={0,1} for
        // lanes 0-15 and K={2,3} for lanes 16-31.
        v2f A[4];
#pragma unroll
        for (int s = 0; s < 4; ++s) {
            const int kk = s * 4 + half * 2;
            A[s].x = (et == kk)     ? val : 0.0f;
            A[s].y = (et == kk + 1) ? val : 0.0f;
        }

        // 16 WMMAs -> 4 accumulators (D[e][nt*16 + l16] tiles).
        v8f accs[4];
#pragma unroll
        for (int nt = 0; nt < 4; ++nt) {
            v8f acc = {};
#pragma unroll
            for (int s = 0; s < 4; ++s)
                acc = __builtin_amdgcn_wmma_f32_16x16x4_f32(
                        false, A[s], false, Bop[s][nt],
                        (short)0, acc, false, false);
            accs[nt] = acc;
        }

        // Scatter. D layout: component i -> edge m = i + 8*half, col l16.
        // One bpermute + one base address per i; 4 atomics with imm offsets.
        if (b * 16 + 16 <= E) {
            // Full batch: no per-element guards, EXEC stays all-1s.
#pragma unroll
            for (int i = 0; i < 8; ++i) {
                const int m  = i + half * 8;
                const int rc = __shfl(rowcol, m, 32);
                float* p = out + ((size_t)(unsigned)rc << 6) + (size_t)l16;
                atomicAdd(p,      accs[0][i]);
                atomicAdd(p + 16, accs[1][i]);
                atomicAdd(p + 32, accs[2][i]);
                atomicAdd(p + 48, accs[3][i]);
            }
        } else {
            // Tail batch: guard each edge's 4 atomics together.
#pragma unroll
            for (int i = 0; i < 8; ++i) {
                const int m  = i + half * 8;
                const int rc = __shfl(rowcol, m, 32);
                if (b * 16 + m < E) {
                    float* p = out + ((size_t)(unsigned)rc << 6) + (size_t)l16;
                    atomicAdd(p,      accs[0][i]);
                    atomicAdd(p + 16, accs[1][i]);
                    atomicAdd(p + 32, accs[2][i]);
                    atomicAdd(p + 48, accs[3][i]);
                }
            }
        }
    }
}

extern "C" void kernel_launch(void* const* d_in, const int* in_sizes, int n_in,
                              void* d_out, int out_size, void* d_ws, size_t ws_size,
                              hipStream_t stream) {
    const int*   edge_type = (const int*)  d_in[0];  // (E,1) int32
    const int2*  edge_out  = (const int2*) d_in[1];  // (E,2) int32
    const float* edge_val  = (const float*)d_in[2];  // (E)   f32
    const float* emb       = (const float*)d_in[3];  // (16,64) f32
    const int E = in_sizes[0];

    // out_size = N*N*64  ->  N
    const int N = (int)(sqrt((double)out_size / 64.0) + 0.5);

    // 1) zero-init 256MB output: grid-stride float4 stores
    const long long n  = (long long)out_size;
    const long long n4 = n >> 2;
    zero_out_kernel<<<4096, 256, 0, stream>>>((float4*)d_out, n4,
                                              (float*)d_out, n);

    // 2) WMMA gather + atomic scatter: one wave per 16-edge batch
    const int nbat   = (E + 15) / 16;
    int blocks = (nbat + 7) / 8;            // 8 waves per 256-thread block
    if (blocks > 2048) blocks = 2048;
    if (blocks < 1)    blocks = 1;
    edge_wmma_scatter<<<blocks, 256, 0, stream>>>(edge_type, edge_out, edge_val,
                                                  emb, (float*)d_out, E, N);
}